// SubjectLayers_67284957659761
// MI455X (gfx1250) — compile-verified
//
#include <hip/hip_runtime.h>

// SubjectLayers: out[b,o,t] = sum_c W[subj[b],o,c] * x[b,c,t] + bias[subj[b],o]
// B=128, C=64, T=8192, S=16, fp32.  Memory-bound (x read once, out written once);
// compute via V_WMMA_F32_16X16X4_F32 (CDNA5 fp32 matrix path).

typedef __attribute__((ext_vector_type(2))) float v2f;
typedef __attribute__((ext_vector_type(8))) float v8f;

#define C_DIM 64
#define T_DIM 8192
#define WAVES_PER_BLOCK 8
#define TCHUNKS 8                      // blocks per sample along T
#define T_PER_CHUNK (T_DIM / TCHUNKS)  // 1024
#define TILES_PER_CHUNK (T_PER_CHUNK / 16)  // 64

__global__ __launch_bounds__(256) void subject_layers_wmma_f32(
    const float* __restrict__ x,     // [B, C, T]
    const int*   __restrict__ subj,  // [B]
    const float* __restrict__ W,     // [S, C, C]
    const float* __restrict__ bias,  // [S, C]
    float* __restrict__ out)         // [B, C, T]
{
    const int blk   = blockIdx.x;
    const int b     = blk / TCHUNKS;
    const int tch   = blk % TCHUNKS;
    const int wave  = threadIdx.x >> 5;
    const int lane  = threadIdx.x & 31;
    const int ln    = lane & 15;   // N column (B/D frags) or M row (A frag)
    const int khalf = lane >> 4;   // K split: lanes 0-15 -> K={0,1}, 16-31 -> K={2,3}

    const int s = subj[b];
    const float* __restrict__ Wb = W + (size_t)s * (C_DIM * C_DIM);
    const float* __restrict__ xb = x + (size_t)b * (C_DIM * T_DIM);
    float* __restrict__       ob = out + (size_t)b * (C_DIM * T_DIM);

    // ---- Preload all A fragments (W tiles) for this subject: 4 M-tiles x 16 K-steps.
    // A(16x4 f32) layout: lane l holds row M=l&15; VGPR0 = K0(+2*khalf), VGPR1 = K0+1(+2*khalf).
    v2f afrag[4][16];
#pragma unroll
    for (int mt = 0; mt < 4; ++mt) {
        const int m = mt * 16 + ln;
#pragma unroll
        for (int kk = 0; kk < 16; ++kk) {
            const int k = kk * 4 + khalf * 2;
            const float2 w2 = *reinterpret_cast<const float2*>(Wb + m * C_DIM + k); // b64 load
            v2f a; a.x = w2.x; a.y = w2.y;
            afrag[mt][kk] = a;
        }
    }

    // ---- Bias, laid out to match the D fragment: lane holds M = mt*16 + khalf*8 + r.
    float biasv[4][8];
#pragma unroll
    for (int mt = 0; mt < 4; ++mt) {
        const int mbase = mt * 16 + khalf * 8;
        const float4 b0 = *reinterpret_cast<const float4*>(bias + s * C_DIM + mbase);
        const float4 b1 = *reinterpret_cast<const float4*>(bias + s * C_DIM + mbase + 4);
        biasv[mt][0] = b0.x; biasv[mt][1] = b0.y; biasv[mt][2] = b0.z; biasv[mt][3] = b0.w;
        biasv[mt][4] = b1.x; biasv[mt][5] = b1.y; biasv[mt][6] = b1.z; biasv[mt][7] = b1.w;
    }

    const int t_base = tch * T_PER_CHUNK;

    // Each wave strides over 16-wide t-tiles of this chunk (uniform loop -> EXEC stays all-1s).
    for (int tile = wave; tile < TILES_PER_CHUNK; tile += WAVES_PER_BLOCK) {
        const int t0 = t_base + tile * 16;

        // ---- Load B fragments (x tile 64x16): lane l holds col N=ln; K split as in A.
        v2f bfrag[16];
#pragma unroll
        for (int kk = 0; kk < 16; ++kk) {
            const int k = kk * 4 + khalf * 2;
            v2f bv;
            bv.x = xb[(size_t)k       * T_DIM + t0 + ln];
            bv.y = xb[(size_t)(k + 1) * T_DIM + t0 + ln];
            bfrag[kk] = bv;
        }

        // ---- 4 output M-tiles; accumulator seeded with bias (constant along N).
#pragma unroll
        for (int mt = 0; mt < 4; ++mt) {
            v8f acc;
#pragma unroll
            for (int r = 0; r < 8; ++r) acc[r] = biasv[mt][r];

#pragma unroll
            for (int kk = 0; kk < 16; ++kk) {
                acc = __builtin_amdgcn_wmma_f32_16x16x4_f32(
                    /*neg_a=*/false, afrag[mt][kk],
                    /*neg_b=*/false, bfrag[kk],
                    /*c_mod=*/(short)0, acc,
                    /*reuse_a=*/false, /*reuse_b=*/false);
            }

            // ---- Store D: lane l -> (M = mt*16 + khalf*8 + r, N = ln).
            const int mbase = mt * 16 + khalf * 8;
#pragma unroll
            for (int r = 0; r < 8; ++r) {
                ob[(size_t)(mbase + r) * T_DIM + t0 + ln] = acc[r];
            }
        }
    }
}

extern "C" void kernel_launch(void* const* d_in, const int* in_sizes, int n_in,
                              void* d_out, int out_size, void* d_ws, size_t ws_size,
                              hipStream_t stream) {
    const float* x    = (const float*)d_in[0];  // [B, C, T]
    const int*   subj = (const int*)  d_in[1];  // [B]
    const float* W    = (const float*)d_in[2];  // [S, C, C]
    const float* bias = (const float*)d_in[3];  // [S, C]
    float*       out  = (float*)d_out;          // [B, C, T]

    const int B = in_sizes[1];                  // subj_idx length
    dim3 grid(B * TCHUNKS);
    dim3 block(32 * WAVES_PER_BLOCK);
    subject_layers_wmma_f32<<<grid, block, 0, stream>>>(x, subj, W, bias, out);
}